// GCN_54322746359932
// MI455X (gfx1250) — compile-verified
//
#include <hip/hip_runtime.h>
#include <math.h>

#define DIN  128
#define DH   128
#define DOUT 64
#define NLH  2   // extra hidden layers

typedef __attribute__((ext_vector_type(2))) float v2f;
typedef __attribute__((ext_vector_type(8))) float v8f;

#if defined(__gfx1250__) && __has_builtin(__builtin_amdgcn_wmma_f32_16x16x4_f32)
#define HAVE_WMMA_F32 1
#else
#define HAVE_WMMA_F32 0
#endif

// ---------------------------------------------------------------------------
// degree / normalization prep
// ---------------------------------------------------------------------------
__global__ void k_deg_init(float* __restrict__ deg, int N) {
  int i = blockIdx.x * blockDim.x + threadIdx.x;
  if (i < N) deg[i] = 1.0f;  // self-loop weight 1
}

__global__ void k_deg_scatter(const int* __restrict__ dst,
                              const float* __restrict__ w,
                              float* __restrict__ deg, int E) {
  int e = blockIdx.x * blockDim.x + threadIdx.x;
  if (e < E) atomicAdd(&deg[dst[e]], w[e]);
}

__global__ void k_rsqrt_inplace(float* __restrict__ d, int N) {
  int i = blockIdx.x * blockDim.x + threadIdx.x;
  if (i < N) d[i] = rsqrtf(d[i]);   // deg >= 1 always (self-loops)
}

__global__ void k_edge_norm(const int* __restrict__ src,
                            const int* __restrict__ dst,
                            const float* __restrict__ w,
                            const float* __restrict__ dinv,
                            float* __restrict__ nrm, int E) {
  int e = blockIdx.x * blockDim.x + threadIdx.x;
  if (e < E) nrm[e] = dinv[src[e]] * w[e] * dinv[dst[e]];
}

// ---------------------------------------------------------------------------
// Tiled fp32 WMMA GEMM:  C[M x NCOL] = A[M x 128] * B[128 x NCOL]
// One block computes a 16-row stripe; wave w owns output cols [16w, 16w+16).
// A stripe staged in LDS (padded to 132 floats/row -> conflict-free).
// Fragment layouts per CDNA5 ISA 7.12.2 (32-bit A 16x4 / B 4x16 / C 16x16).
// ---------------------------------------------------------------------------
template <int NCOL, bool BIAS_RELU>
__global__ __launch_bounds__(32 * (NCOL / 16))
void k_gemm_wmma(const float* __restrict__ A, const float* __restrict__ B,
                 const float* __restrict__ bias, float* __restrict__ C, int M) {
  constexpr int K = 128;
  constexpr int NTHREADS = 32 * (NCOL / 16);
  __shared__ float As[16 * 132];

  const int m0  = blockIdx.x * 16;
  const int tid = threadIdx.x;

  // cooperative load of the 16 x 128 A stripe (float4 granularity)
  for (int idx = tid; idx < 16 * (K / 4); idx += NTHREADS) {
    int r  = idx >> 5;        // 32 float4 per row
    int c4 = idx & 31;
    int row = m0 + r;
    float4 v = (row < M) ? ((const float4*)(A + (size_t)row * K))[c4]
                         : make_float4(0.f, 0.f, 0.f, 0.f);
    float* p = &As[r * 132 + c4 * 4];
    p[0] = v.x; p[1] = v.y; p[2] = v.z; p[3] = v.w;
  }
  __syncthreads();

  const int wave    = tid >> 5;
  const int lane    = tid & 31;
  const int colbase = wave * 16;
  const int hi      = lane >> 4;   // lane half-group
  const int lr      = lane & 15;

  v8f acc = {};
#if HAVE_WMMA_F32
  for (int k0 = 0; k0 < K; k0 += 4) {
    const int ka = k0 + hi * 2;
    v2f a, b;
    // A 16x4: lanes 0-15 -> rows, VGPR0/1 = K {ka, ka+1}, lanes 16-31 offset K by 2
    a.x = As[lr * 132 + ka];
    a.y = As[lr * 132 + ka + 1];
    // B 4x16: lanes -> cols, VGPR0/1 = K {ka, ka+1}
    b.x = B[(size_t)ka * NCOL + colbase + lr];
    b.y = B[(size_t)(ka + 1) * NCOL + colbase + lr];
    acc = __builtin_amdgcn_wmma_f32_16x16x4_f32(
        /*neg_a=*/false, a, /*neg_b=*/false, b,
        /*c_mod=*/(short)0, acc, /*reuse_a=*/false, /*reuse_b=*/false);
  }
#else
  // scalar fallback with identical output mapping
  for (int r = 0; r < 8; ++r) {
    float s = 0.f;
    const int row = r + hi * 8;
    for (int k = 0; k < K; ++k)
      s += As[row * 132 + k] * B[(size_t)k * NCOL + colbase + lr];
    acc[r] = s;
  }
#endif

  // C 16x16: vgpr r -> row r (+8 for upper lane half), col = lane&15
#pragma unroll
  for (int r = 0; r < 8; ++r) {
    const int row = m0 + r + hi * 8;
    if (row < M) {
      float v = acc[r];
      if (BIAS_RELU) {
        v += bias[colbase + lr];
        v = v > 0.f ? v : 0.f;
      }
      C[(size_t)row * NCOL + colbase + lr] = v;
    }
  }
}

// ---------------------------------------------------------------------------
// aggregation
// ---------------------------------------------------------------------------
// self-loop contribution initializes the accumulator (no atomics needed)
__global__ void k_selfloop_init(const float* __restrict__ t,
                                const float* __restrict__ dinv,
                                float* __restrict__ out, int N) {
  int idx = blockIdx.x * blockDim.x + threadIdx.x;
  if (idx < N * DH) {
    int i = idx >> 7;             // DH = 128
    float d = dinv[i];
    out[idx] = d * d * t[idx];
  }
}

// one wave32 per edge; lanes stripe the 128 features (float4 each)
__global__ void k_edge_scatter(const int* __restrict__ src,
                               const int* __restrict__ dst,
                               const float* __restrict__ nrm,
                               const float* __restrict__ t,
                               float* __restrict__ out, int E) {
  int e    = blockIdx.x * (blockDim.x >> 5) + (threadIdx.x >> 5);
  int lane = threadIdx.x & 31;
  if (e >= E) return;
  const int s = src[e];
  const int d = dst[e];
  const float n = nrm[e];
  float4 v = ((const float4*)(t + (size_t)s * DH))[lane];
  float* o = out + (size_t)d * DH + lane * 4;
  atomicAdd(o + 0, n * v.x);
  atomicAdd(o + 1, n * v.y);
  atomicAdd(o + 2, n * v.z);
  atomicAdd(o + 3, n * v.w);
}

// out = elu(out + bias), in place
__global__ void k_bias_elu(float* __restrict__ h, const float* __restrict__ bias,
                           int N) {
  int idx = blockIdx.x * blockDim.x + threadIdx.x;
  if (idx < N * DH) {
    float v = h[idx] + bias[idx & (DH - 1)];
    h[idx] = v > 0.f ? v : (__expf(v) - 1.f);
  }
}

// ---------------------------------------------------------------------------
extern "C" void kernel_launch(void* const* d_in, const int* in_sizes, int n_in,
                              void* d_out, int out_size, void* d_ws, size_t ws_size,
                              hipStream_t stream) {
  const float* x   = (const float*)d_in[0];
  const int*   ei  = (const int*)d_in[1];
  const float* ew  = (const float*)d_in[2];
  const float* W1  = (const float*)d_in[3];
  const float* b1  = (const float*)d_in[4];
  const float* Wh  = (const float*)d_in[5];
  const float* bh  = (const float*)d_in[6];
  const float* Wo  = (const float*)d_in[7];
  const float* bo  = (const float*)d_in[8];

  const int N = in_sizes[0] / DIN;
  const int E = in_sizes[2];
  const int* src = ei;
  const int* dst = ei + E;

  float* ws   = (float*)d_ws;
  float* dinv = ws;                           // N   (deg then rsqrt, in place)
  float* nrm  = dinv + N;                     // E
  float* hA   = nrm + E;                      // N*DH
  float* hB   = hA + (size_t)N * DH;          // N*DH
  float* t    = hB + (size_t)N * DH;          // N*DH

  const int TPB = 256;
  const int gN  = (N + TPB - 1) / TPB;
  const int gE  = (E + TPB - 1) / TPB;
  const int gNF = (N * DH + TPB - 1) / TPB;
  const int gEd = (E + (TPB / 32) - 1) / (TPB / 32);   // wave per edge
  const int gM  = (N + 15) / 16;

  // --- normalization coefficients ---
  k_deg_init<<<gN, TPB, 0, stream>>>(dinv, N);
  k_deg_scatter<<<gE, TPB, 0, stream>>>(dst, ew, dinv, E);
  k_rsqrt_inplace<<<gN, TPB, 0, stream>>>(dinv, N);
  k_edge_norm<<<gE, TPB, 0, stream>>>(src, dst, ew, dinv, nrm, E);

  // --- layer 1: t = x @ W1 ; hA = elu(aggregate(t) + b1) ---
  k_gemm_wmma<DH, false><<<gM, 256, 0, stream>>>(x, W1, nullptr, t, N);
  k_selfloop_init<<<gNF, TPB, 0, stream>>>(t, dinv, hA, N);
  k_edge_scatter<<<gEd, TPB, 0, stream>>>(src, dst, nrm, t, hA, E);
  k_bias_elu<<<gNF, TPB, 0, stream>>>(hA, b1, N);

  // --- hidden layers ---
  float* cur = hA;
  float* nxt = hB;
  for (int l = 0; l < NLH; ++l) {
    const float* W = Wh + (size_t)l * DH * DH;
    const float* b = bh + (size_t)l * DH;
    k_gemm_wmma<DH, false><<<gM, 256, 0, stream>>>(cur, W, nullptr, t, N);
    k_selfloop_init<<<gNF, TPB, 0, stream>>>(t, dinv, nxt, N);
    k_edge_scatter<<<gEd, TPB, 0, stream>>>(src, dst, nrm, t, nxt, E);
    k_bias_elu<<<gNF, TPB, 0, stream>>>(nxt, b, N);
    float* tmp = cur; cur = nxt; nxt = tmp;
  }

  // --- output layer: d_out = relu(cur @ Wo + bo), fused epilogue ---
  k_gemm_wmma<DOUT, true><<<gM, 128, 0, stream>>>(cur, Wo, bo, (float*)d_out, N);
}